// TDTLossNumba_48919677501622
// MI455X (gfx1250) — compile-verified
//
#include <hip/hip_runtime.h>
#include <math.h>

#define BLANKC 512
#define SIGMA  0.05f
#define NEGV   (-1e30f)
#define Bn 8
#define Tn 256
#define Un 65
#define Vn 513
#define VD 518            // V + N_DUR

typedef __attribute__((ext_vector_type(2))) float v2f;
typedef __attribute__((ext_vector_type(8))) float v8f;
typedef __attribute__((ext_vector_type(4))) unsigned int v4u;
typedef __attribute__((ext_vector_type(4))) int v4i;
typedef __attribute__((ext_vector_type(8))) int v8i;

#if defined(__has_builtin)
#if __has_builtin(__builtin_amdgcn_tensor_load_to_lds) && \
    __has_builtin(__builtin_amdgcn_s_wait_tensorcnt)
#define HAVE_TDM 1
#else
#define HAVE_TDM 0
#endif
#else
#define HAVE_TDM 0
#endif

__device__ __forceinline__ float lap(float x, float y) {     // logaddexp
  float m = fmaxf(x, y);
  float d = fminf(x, y) - m;                                 // <= 0
  return m + log1pf(expf(d));
}
__device__ __forceinline__ float lse4(float a, float b, float c, float d) {
  float m = fmaxf(fmaxf(a, b), fmaxf(c, d));
  float s = expf(a - m) + expf(b - m) + expf(c - m) + expf(d - m);
  return m + logf(s);
}

// ---------------------------------------------------------------------------
// Kernel A: streaming log-softmax extraction. One wave per (b, u, 16 t-rows).
// The 16x518-float strided tile is DMA'd into LDS by the Tensor Data Mover
// (one TENSOR_LOAD_TO_LDS + s_wait_tensorcnt), then row-max via VALU+shfl and
// exp-sum accumulated on the matrix pipe with V_WMMA_F32_16X16X4_F32 (B=ones).
// ---------------------------------------------------------------------------
__global__ __launch_bounds__(32)
void tdt_prep_kernel(const float* __restrict__ acts,
                     const int*   __restrict__ labels,
                     float* __restrict__ ws_blank,
                     float* __restrict__ ws_emit,
                     float* __restrict__ ws_dur) {
  // 16 rows x 518 floats, contiguous (TDM writes tile row-major, no padding),
  // + 16 floats for the WMMA row-sum exchange.
  __shared__ __align__(16) float smem[16 * VD + 16];
  const int lane = threadIdx.x;
  const int t0 = blockIdx.x * 16;
  const int u  = blockIdx.y;
  const int b  = blockIdx.z;

#if HAVE_TDM
  {
    // ---- Tensor DMA descriptor (D#), ISA 8.3-8.6 ----
    size_t ga = (size_t)acts + ((size_t)((b * Tn + t0) * Un + u)) * VD * 4;
    unsigned lds_addr = (unsigned)(size_t)&smem[0];   // flat[31:0] == LDS addr
    v4u g0;
    g0[0] = 1u;                                       // count=1, user descriptor
    g0[1] = lds_addr;                                 // lds_addr
    g0[2] = (unsigned)(ga & 0xFFFFFFFFu);             // global_addr[31:0]
    g0[3] = (unsigned)((ga >> 32) & 0x01FFFFFFu)      // global_addr[56:32]
            | 0x80000000u;                            // type = 2 ("image")
    v8i g1;
    g1[0] = (int)(2u << 16);          // workgroup_mask=0, data_size=4B
    g1[1] = (int)((unsigned)VD << 16);// tensor_dim0 = 518 (in [63:48])
    g1[2] = (int)(16u << 16);         // tensor_dim0 hi=0, tensor_dim1 = 16
    g1[3] = (int)((unsigned)VD << 16);// tensor_dim1 hi=0, tile_dim0 = 518
    g1[4] = 16;                       // tile_dim1 = 16, tile_dim2 = 0 (2D)
    g1[5] = Un * VD;                  // tensor_dim0_stride = 33670 elements
    g1[6] = 0;                        // stride hi / tensor_dim1_stride lo
    g1[7] = 0;                        // tensor_dim1_stride hi
    v4i gz; gz[0] = 0; gz[1] = 0; gz[2] = 0; gz[3] = 0;
#if __clang_major__ >= 23
    v8i gz8; gz8[0]=0; gz8[1]=0; gz8[2]=0; gz8[3]=0;
             gz8[4]=0; gz8[5]=0; gz8[6]=0; gz8[7]=0;
    __builtin_amdgcn_tensor_load_to_lds(g0, g1, gz, gz, gz8, 0);
#else
    __builtin_amdgcn_tensor_load_to_lds(g0, g1, gz, gz, 0);
#endif
    __builtin_amdgcn_s_wait_tensorcnt((short)0);      // TENSORcnt == 0
  }
  __syncthreads();
#else
  // fallback: cooperative load/store staging (row pitch 259 float2 = 518 f)
  {
    float2* smem2 = reinterpret_cast<float2*>(smem);
    const float2* g2 = reinterpret_cast<const float2*>(acts);
    for (int i = lane; i < 16 * 259; i += 32) {
      int row = i / 259;
      int c2  = i - row * 259;
      size_t gi = ((size_t)((b * Tn + t0 + row) * Un + u)) * 259 + c2;
      if ((i & 15) == 0) __builtin_prefetch(g2 + gi + 512, 0, 1);
      smem2[row * 259 + c2] = g2[gi];
    }
  }
  __syncthreads();
#endif

  const int r  = lane & 15;   // row in tile (= WMMA M index)
  const int h  = lane >> 4;   // half selects K columns {0,1} vs {2,3}
  const int rb = r * VD;

  // ---- pass 1: row max over 513 label logits ----
  float m = -3.4e38f;
  #pragma unroll 8
  for (int k0 = 0; k0 < 512; k0 += 4) {
    float2 v = *reinterpret_cast<const float2*>(&smem[rb + k0 + 2 * h]);
    m = fmaxf(m, fmaxf(v.x, v.y));
  }
  if (h == 0) m = fmaxf(m, smem[rb + 512]);
  m = fmaxf(m, __shfl_xor(m, 16, 32));          // full-row max on all lanes

  // ---- pass 2: sum(exp(x - m)) via WMMA: D[r][*] += sum_k A[r][k] * 1 ----
  v8f acc = {0.f, 0.f, 0.f, 0.f, 0.f, 0.f, 0.f, 0.f};
  v2f bones; bones[0] = 1.0f; bones[1] = 1.0f;  // B (4x16) = ones
  #pragma unroll 4
  for (int k0 = 0; k0 < 512; k0 += 4) {
    float2 v = *reinterpret_cast<const float2*>(&smem[rb + k0 + 2 * h]);
    v2f a; a[0] = expf(v.x - m); a[1] = expf(v.y - m);
    acc = __builtin_amdgcn_wmma_f32_16x16x4_f32(false, a, false, bones,
                                                (short)0, acc, false, false);
  }
  { // epilogue: only column 512 remains valid (cols 513.. are duration logits)
    v2f a; a[0] = (h == 0) ? expf(smem[rb + 512] - m) : 0.0f; a[1] = 0.0f;
    acc = __builtin_amdgcn_wmma_f32_16x16x4_f32(false, a, false, bones,
                                                (short)0, acc, false, false);
  }

  // D layout: VGPR j -> row j (lanes 0-15) / row j+8 (lanes 16-31); all N equal
  float* sums = smem + 16 * VD;
  if (lane == 0) {
    #pragma unroll
    for (int j = 0; j < 8; ++j) sums[j] = acc[j];
  }
  if (lane == 16) {
    #pragma unroll
    for (int j = 0; j < 8; ++j) sums[8 + j] = acc[j];
  }
  __syncthreads();

  // ---- outputs: lane = row ----
  if (lane < 16) {
    int t = t0 + lane;
    int rowb = lane * VD;
    float lse = m + logf(sums[lane]);
    size_t o = (size_t)(b * Tn + t) * Un + u;
    ws_blank[o] = smem[rowb + BLANKC] - lse - SIGMA;
    if (u < Un - 1) {
      int lab = labels[b * (Un - 1) + u];
      ws_emit[(size_t)(b * Tn + t) * (Un - 1) + u] = smem[rowb + lab] - lse - SIGMA;
    }
    float d0 = smem[rowb + 513], d1 = smem[rowb + 514], d2 = smem[rowb + 515],
          d3 = smem[rowb + 516], d4 = smem[rowb + 517];
    float dm = fmaxf(fmaxf(fmaxf(d0, d1), fmaxf(d2, d3)), d4);
    float ds = expf(d0 - dm) + expf(d1 - dm) + expf(d2 - dm) +
               expf(d3 - dm) + expf(d4 - dm);
    float dl = dm + logf(ds);
    float* dp = ws_dur + o * 5;
    dp[0] = d0 - dl; dp[1] = d1 - dl; dp[2] = d2 - dl;
    dp[3] = d3 - dl; dp[4] = d4 - dl;
  }
}

// ---------------------------------------------------------------------------
// Kernel B: forward DP. One wave per batch. Lane l owns u = 2l+1, 2l+2.
// The serial u-recurrence a_u = logaddexp(base_u, a_{u-1}+e_u) is run as a
// log-semiring affine-map parallel scan (5 shfl_up compose steps).
// Alpha history for t-1..t-4 kept in registers; rows also stored to ws.
// ---------------------------------------------------------------------------
__global__ __launch_bounds__(32)
void tdt_dp_kernel(const float* __restrict__ ws_blank,
                   const float* __restrict__ ws_emit,
                   const float* __restrict__ ws_dur,
                   const int*   __restrict__ act_lens,
                   const int*   __restrict__ label_lens,
                   float* __restrict__ ws_alpha,
                   float* __restrict__ ws_cost) {
  const int b    = blockIdx.x;
  const int lane = threadIdx.x;
  const int u1  = 2 * lane + 1;
  const int u2  = 2 * lane + 2;
  const int up0 = 2 * lane;       // emit index u-1 for map u1
  const int up1 = 2 * lane + 1;   // emit index u-1 for map u2

  // history: h0 = alpha[t-j-1][2l] (lane0: u=0), h1 = [2l+1], h2 = [2l+2]
  float h0[4], h1[4], h2[4];
  #pragma unroll
  for (int j = 0; j < 4; ++j) { h0[j] = NEGV; h1[j] = NEGV; h2[j] = NEGV; }

  for (int t = 0; t < Tn; ++t) {
    {
      // prefetch rows of time t that next iteration consumes at d=1
      size_t rp = (size_t)(b * Tn + t) * Un;
      __builtin_prefetch(ws_blank + rp + u1, 0, 1);
      __builtin_prefetch(ws_dur + (rp + u1) * 5, 0, 1);
    }
    float tne1[4], tne2[4], tem0[4], tem1[4], tne0[4];
    #pragma unroll
    for (int j = 0; j < 4; ++j) {
      int d = j + 1;
      if (t - d >= 0) {
        size_t rb = (size_t)(b * Tn + (t - d)) * Un;
        size_t re = (size_t)(b * Tn + (t - d)) * (Un - 1);
        tne1[j] = h1[j] + ws_blank[rb + u1] + ws_dur[(rb + u1) * 5 + d];
        tne2[j] = h2[j] + ws_blank[rb + u2] + ws_dur[(rb + u2) * 5 + d];
        tem0[j] = h0[j] + ws_emit[re + up0] + ws_dur[(rb + up0) * 5 + d];
        tem1[j] = h1[j] + ws_emit[re + up1] + ws_dur[(rb + up1) * 5 + d];
        tne0[j] = h0[j] + ws_blank[rb]      + ws_dur[rb * 5 + d]; // lane 0 only
      } else {
        tne1[j] = NEGV; tne2[j] = NEGV; tem0[j] = NEGV; tem1[j] = NEGV; tne0[j] = NEGV;
      }
    }
    float ne1 = lse4(tne1[0], tne1[1], tne1[2], tne1[3]);
    float ne2 = lse4(tne2[0], tne2[1], tne2[2], tne2[3]);
    float em0 = lse4(tem0[0], tem0[1], tem0[2], tem0[3]);
    float em1 = lse4(tem1[0], tem1[1], tem1[2], tem1[3]);
    float ne0 = lse4(tne0[0], tne0[1], tne0[2], tne0[3]);
    ne0 = __shfl(ne0, 0, 32);                       // lane 0 holds true u=0 value

    size_t rt = (size_t)(b * Tn + t) * Un;
    size_t re = (size_t)(b * Tn + t) * (Un - 1);
    float e01 = ws_emit[re + up0] + ws_dur[(rt + up0) * 5 + 0];
    float e02 = ws_emit[re + up1] + ws_dur[(rt + up1) * 5 + 0];
    float base1 = lap(ne1, em0);
    float base2 = lap(ne2, em1);
    float a0 = (t == 0) ? 0.0f : ne0;

    // per-lane composed map P = g2 o g1 (each g applied as x -> lap(b, x + e))
    float Sb = lap(base2, base1 + e02);
    float Se = e01 + e02;
    // inclusive scan: S_l = P_l o ... o P_0
    #pragma unroll
    for (int off = 1; off < 32; off <<= 1) {
      float pb = __shfl_up(Sb, off, 32);
      float pe = __shfl_up(Se, off, 32);
      if (lane >= off) {
        Sb = lap(Sb, pb + Se);   // uses old Se
        Se = pe + Se;
      }
    }
    float Spb = __shfl_up(Sb, 1, 32);
    float Spe = __shfl_up(Se, 1, 32);
    if (lane == 0) { Spb = NEGV; Spe = 0.0f; }      // identity map
    float T1b = lap(base1, Spb + e01);
    float T1e = Spe + e01;
    float a1 = lap(T1b, a0 + T1e);                  // alpha[t][2l+1]
    float a2 = lap(Sb,  a0 + Se);                   // alpha[t][2l+2]

    ws_alpha[rt + u1] = a1;
    ws_alpha[rt + u2] = a2;
    if (lane == 0) ws_alpha[rt] = a0;

    #pragma unroll
    for (int j = 3; j > 0; --j) { h0[j] = h0[j-1]; h1[j] = h1[j-1]; h2[j] = h2[j-1]; }
    float nh0 = __shfl_up(a2, 1, 32);               // alpha[t][2l] from lane l-1
    if (lane == 0) nh0 = a0;
    h0[0] = nh0; h1[0] = a1; h2[0] = a2;
  }

  if (lane == 0) {
    int ui = label_lens[b];
    int al = act_lens[b];
    float v[4];
    #pragma unroll
    for (int j = 0; j < 4; ++j) {
      int d = j + 1;
      int ts = al - d;
      if (ts >= 0) {
        size_t rb = (size_t)(b * Tn + ts) * Un;
        v[j] = ws_alpha[rb + ui] + ws_blank[rb + ui] + ws_dur[(rb + ui) * 5 + d];
      } else {
        v[j] = NEGV;
      }
    }
    ws_cost[b] = -lse4(v[0], v[1], v[2], v[3]);
  }
}

// ---------------------------------------------------------------------------
// Kernel C: final mean (deterministic, no atomics)
// ---------------------------------------------------------------------------
__global__ void tdt_final_kernel(const float* __restrict__ ws_cost,
                                 float* __restrict__ out) {
  float v = (threadIdx.x < Bn) ? ws_cost[threadIdx.x] : 0.0f;
  v += __shfl_xor(v, 4, 32);
  v += __shfl_xor(v, 2, 32);
  v += __shfl_xor(v, 1, 32);
  if (threadIdx.x == 0) out[0] = v * (1.0f / (float)Bn);
}

extern "C" void kernel_launch(void* const* d_in, const int* in_sizes, int n_in,
                              void* d_out, int out_size, void* d_ws, size_t ws_size,
                              hipStream_t stream) {
  const float* acts       = (const float*)d_in[0];
  const int*   labels     = (const int*)d_in[1];
  const int*   act_lens   = (const int*)d_in[2];
  const int*   label_lens = (const int*)d_in[3];

  float* ws       = (float*)d_ws;
  float* ws_blank = ws;                                     // B*T*U
  float* ws_emit  = ws_blank + (size_t)Bn * Tn * Un;        // B*T*(U-1)
  float* ws_dur   = ws_emit  + (size_t)Bn * Tn * (Un - 1);  // B*T*U*5
  float* ws_alpha = ws_dur   + (size_t)Bn * Tn * Un * 5;    // B*T*U
  float* ws_cost  = ws_alpha + (size_t)Bn * Tn * Un;        // B

  dim3 gridA(Tn / 16, Un, Bn);
  tdt_prep_kernel<<<gridA, 32, 0, stream>>>(acts, labels, ws_blank, ws_emit, ws_dur);
  tdt_dp_kernel<<<Bn, 32, 0, stream>>>(ws_blank, ws_emit, ws_dur,
                                       act_lens, label_lens, ws_alpha, ws_cost);
  tdt_final_kernel<<<1, 32, 0, stream>>>(ws_cost, (float*)d_out);
}